// NonbondedTerm_45191645888923
// MI455X (gfx1250) — compile-verified
//
#include <hip/hip_runtime.h>
#include <hip/hip_bf16.h>

// MI455X (gfx1250) nonbonded pair-energy reduction.
// Bandwidth-bound: 24 streamed bytes/pair from HBM (NT-hinted), coord gathers
// are L2-resident (1.2 MB table vs 192 MB L2). Wave32 cross-lane reduction is
// done with v_wmma_f32_16x16x4_f32 (B = ones => row sums) + one xor-16 shuffle.

typedef __attribute__((ext_vector_type(2))) float v2f;
typedef __attribute__((ext_vector_type(8))) float v8f;

// Sum a float across all 32 lanes of the wave using one f32 WMMA.
// A (16x4): lane L<16 -> A[L][0]=x, A[L][1]=0 ; lane L>=16 -> A[L-16][2]=x, A[L-16][3]=0.
// B = all ones (layout-independent). D[m][n] = x_m + x_{m+16} for every n.
// Each lane sums its 8 D VGPRs (half-wave total), then xor-16 add completes it.
__device__ __forceinline__ float wave_sum_wmma(float x) {
    v2f a; a[0] = x;    a[1] = 0.0f;
    v2f b; b[0] = 1.0f; b[1] = 1.0f;
    v8f c = {};
    c = __builtin_amdgcn_wmma_f32_16x16x4_f32(
        /*neg_a=*/false, a, /*neg_b=*/false, b,
        /*c_mod=*/(short)0, c, /*reuse_a=*/false, /*reuse_b=*/false);
    float s = ((c[0] + c[1]) + (c[2] + c[3])) + ((c[4] + c[5]) + (c[6] + c[7]));
    s += __shfl_xor(s, 16, 32);
    return s;  // every lane holds the 32-lane total
}

__global__ __launch_bounds__(256) void pair_energy_partial(
    const float* __restrict__ coords,
    const int*   __restrict__ ii,
    const int*   __restrict__ jj,
    const float* __restrict__ coul,
    const float* __restrict__ a12,
    const float* __restrict__ b6,
    const float* __restrict__ b10,
    int npair,
    float* __restrict__ partial)   // [gridDim.x * 2] : (e_coul, e_vdw) per block
{
    float ec = 0.0f, ev = 0.0f;
    const int stride = gridDim.x * blockDim.x;
    for (int p = blockIdx.x * blockDim.x + threadIdx.x; p < npair; p += stride) {
        // Read-once streams: non-temporal so they don't evict the coord table.
        const int   i   = __builtin_nontemporal_load(&ii[p]);
        const int   j   = __builtin_nontemporal_load(&jj[p]);
        const float q   = __builtin_nontemporal_load(&coul[p]);
        const float A   = __builtin_nontemporal_load(&a12[p]);
        const float B6  = __builtin_nontemporal_load(&b6[p]);
        const float B10 = __builtin_nontemporal_load(&b10[p]);

        // L2-resident gathers (coords table = 1.2 MB).
        const float* ci = coords + 3 * i;
        const float* cj = coords + 3 * j;
        const float dx = ci[0] - cj[0];
        const float dy = ci[1] - cj[1];
        const float dz = ci[2] - cj[2];

        const float r2     = dx * dx + dy * dy + dz * dz;
        const float inv_r2 = 1.0f / r2;
        const float inv_r  = sqrtf(inv_r2);
        const float inv_r6 = inv_r2 * inv_r2 * inv_r2;

        ec += q * inv_r;
        ev += A * inv_r6 * inv_r6 - B6 * inv_r6 - B10 * inv_r6 * inv_r2 * inv_r2;
    }

    // Wave32 reduction via WMMA (all 256 threads converged here: EXEC all-1s).
    const float wec = wave_sum_wmma(ec);
    const float wev = wave_sum_wmma(ev);

    __shared__ float sdata[2][8];
    const int lane = threadIdx.x & 31;
    const int wave = threadIdx.x >> 5;
    if (lane == 0) { sdata[0][wave] = wec; sdata[1][wave] = wev; }
    __syncthreads();
    if (threadIdx.x == 0) {
        float s0 = 0.0f, s1 = 0.0f;
        #pragma unroll
        for (int w = 0; w < 8; ++w) { s0 += sdata[0][w]; s1 += sdata[1][w]; }
        partial[blockIdx.x * 2 + 0] = s0;
        partial[blockIdx.x * 2 + 1] = s1;
    }
}

// Deterministic final reduction: fixed tree, single block.
__global__ __launch_bounds__(256) void reduce_partials(
    const float* __restrict__ p0, int n0,
    const float* __restrict__ p1, int n1,
    float* __restrict__ out)       // out[4] = {e_coul, e_lj, e_coul14, e_lj14}
{
    __shared__ float sdata[256];
    for (int ch = 0; ch < 4; ++ch) {
        const float* src = (ch < 2) ? p0 : p1;
        const int    n   = (ch < 2) ? n0 : n1;
        const int    off = ch & 1;
        float s = 0.0f;
        for (int b = threadIdx.x; b < n; b += 256) s += src[2 * b + off];
        sdata[threadIdx.x] = s;
        __syncthreads();
        for (int step = 128; step > 0; step >>= 1) {
            if (threadIdx.x < step) sdata[threadIdx.x] += sdata[threadIdx.x + step];
            __syncthreads();
        }
        if (threadIdx.x == 0) out[ch] = sdata[0];
        __syncthreads();
    }
}

extern "C" void kernel_launch(void* const* d_in, const int* in_sizes, int n_in,
                              void* d_out, int out_size, void* d_ws, size_t ws_size,
                              hipStream_t stream) {
    const float* coords = (const float*)d_in[0];

    const int*   pi    = (const int*)  d_in[1];
    const int*   pj    = (const int*)  d_in[2];
    const float* pq    = (const float*)d_in[3];
    const float* pa12  = (const float*)d_in[4];
    const float* pb6   = (const float*)d_in[5];
    const float* pb10  = (const float*)d_in[6];

    const int*   qi    = (const int*)  d_in[7];
    const int*   qj    = (const int*)  d_in[8];
    const float* qq    = (const float*)d_in[9];
    const float* qa12  = (const float*)d_in[10];
    const float* qb6   = (const float*)d_in[11];
    const float* qb10  = (const float*)d_in[12];

    const int npair   = in_sizes[1];
    const int npair14 = in_sizes[7];

    const int B0  = 4096;  // 1M threads, ~8 pairs each for the 8M stream
    const int B14 = 512;   // 131K threads for the 500K stream

    float* part0  = (float*)d_ws;        // B0  * 2 floats
    float* part14 = part0 + (size_t)B0 * 2;  // B14 * 2 floats

    pair_energy_partial<<<B0, 256, 0, stream>>>(
        coords, pi, pj, pq, pa12, pb6, pb10, npair, part0);
    pair_energy_partial<<<B14, 256, 0, stream>>>(
        coords, qi, qj, qq, qa12, qb6, qb10, npair14, part14);
    reduce_partials<<<1, 256, 0, stream>>>(
        part0, B0, part14, B14, (float*)d_out);
}